// GConv_5471788335164
// MI455X (gfx1250) — compile-verified
//
#include <hip/hip_runtime.h>
#include <hip/hip_bf16.h>

#define N_NODES 100000
#define N_EDGES 1600000
#define N_GRAPHS 512
#define DIM 128
#define N_LAYERS 3
#define OUT_DIM (DIM * N_LAYERS)   // 384
#define BN_EPS 1e-5f
#define LDS_PITCH 132              // 16-row A strip pitch -> 64-bank-conflict-free b64 reads

typedef __attribute__((ext_vector_type(2))) float v2f;
typedef __attribute__((ext_vector_type(8))) float v8f;

// ---------------------------------------------------------------- utilities
__global__ void zero_f32(float* __restrict__ p, int n) {
    int i = blockIdx.x * blockDim.x + threadIdx.x;
    if (i < n) p[i] = 0.0f;
}

// u[n][k] = (1 + eps) * z[n][k]    (seeds the aggregation buffer)
__global__ void init_u(float* __restrict__ u, const float* __restrict__ z,
                       int zstride, const float* __restrict__ eps_p) {
    int idx4 = blockIdx.x * blockDim.x + threadIdx.x;   // < N_NODES * 32
    int row  = idx4 >> 5;
    int c    = (idx4 & 31) << 2;
    float s  = 1.0f + *eps_p;
    float4 v = *(const float4*)(z + (size_t)row * zstride + c);
    v.x *= s; v.y *= s; v.z *= s; v.w *= s;
    *(float4*)(u + (size_t)row * DIM + c) = v;
}

// u[dst] += z[src] : one thread per (edge, 4-feature chunk)
__global__ void scatter_edges(float* __restrict__ u,
                              const float* __restrict__ z, int zstride,
                              const int* __restrict__ src,
                              const int* __restrict__ dst) {
    long long t = (long long)blockIdx.x * blockDim.x + threadIdx.x; // < N_EDGES*32
    int e = (int)(t >> 5);
    int c = ((int)t & 31) << 2;
    int s = src[e];
    int d = dst[e];
    float4 v = *(const float4*)(z + (size_t)s * zstride + c);
    float* up = u + (size_t)d * DIM + c;
    atomicAdd(up + 0, v.x);
    atomicAdd(up + 1, v.y);
    atomicAdd(up + 2, v.z);
    atomicAdd(up + 3, v.w);
}

// h = relu(u @ W^T + b); write into z_cat slice (row stride 384);
// accumulate per-feature sum / sumsq for BN.
// Block = 256 threads = 8 waves; wave w owns output columns [16w, 16w+16).
// Each block processes `strips` strips of 16 rows; B fragments live in VGPRs.
__launch_bounds__(256)
__global__ void gemm_bias_relu_stats(const float* __restrict__ u,
                                     const float* __restrict__ Wl,  // [128][128] row-major
                                     const float* __restrict__ bl,  // [128]
                                     float* __restrict__ out,       // z_cat + layer col, stride 384
                                     float* __restrict__ ssum,
                                     float* __restrict__ ssq,
                                     int strips) {
    __shared__ float lds_u[16 * LDS_PITCH];
    __shared__ float lds_sum[DIM];
    __shared__ float lds_sq[DIM];

    const int tid  = threadIdx.x;
    const int wave = tid >> 5;
    const int lane = tid & 31;
    const int half = lane >> 4;     // 0: K pair {0,1} / rows 0..7 ; 1: K pair {2,3} / rows 8..15
    const int nl   = lane & 15;     // A: M index ; B,C,D: N index
    const int j    = wave * 16 + nl; // global output feature

    // ---- preload B fragments for all 32 K-chunks (64 VGPRs, reused every strip)
    float bx[32], by[32];
#pragma unroll
    for (int c = 0; c < 32; ++c) {
        int kb = c * 4 + half * 2;
        bx[c] = Wl[j * DIM + kb];      // B[k][n] = W[n][k] (h = u @ W^T)
        by[c] = Wl[j * DIM + kb + 1];
    }
    const float bias = bl[j];

    float accsum = 0.0f, accsq = 0.0f;
    const int strip0 = blockIdx.x * strips;

    for (int s = 0; s < strips; ++s) {
        const int row0 = (strip0 + s) * 16;
        // cooperative coalesced load of the 16x128 A strip into LDS
        const float4* gsrc = (const float4*)(u + (size_t)row0 * DIM);
#pragma unroll
        for (int r = 0; r < 2; ++r) {
            int idx4 = tid + r * 256;          // 0..511 float4's
            int row  = idx4 >> 5;
            int col  = (idx4 & 31) << 2;
            *(float4*)&lds_u[row * LDS_PITCH + col] = gsrc[idx4];
        }
        __syncthreads();

        v8f acc = {};
#pragma unroll
        for (int c = 0; c < 32; ++c) {
            int kb = c * 4 + half * 2;
            v2f a;
            a.x = lds_u[nl * LDS_PITCH + kb];
            a.y = lds_u[nl * LDS_PITCH + kb + 1];
            v2f bf; bf.x = bx[c]; bf.y = by[c];
            acc = __builtin_amdgcn_wmma_f32_16x16x4_f32(
                      false, a, false, bf, (short)0, acc, false, false);
        }

        // bias + relu + store + per-lane stats (lane covers rows m = v + 8*half, col j)
#pragma unroll
        for (int v = 0; v < 8; ++v) {
            int m = v + half * 8;
            float val = acc[v] + bias;
            val = val > 0.0f ? val : 0.0f;
            out[(size_t)(row0 + m) * OUT_DIM + j] = val;
            accsum += val;
            accsq  += val * val;
        }
        __syncthreads();   // protect lds_u before next strip's stores
    }

    // ---- reduce BN stats: LDS partials, then one global atomic per feature
    if (tid < DIM) { lds_sum[tid] = 0.0f; lds_sq[tid] = 0.0f; }
    __syncthreads();
    atomicAdd(&lds_sum[j], accsum);
    atomicAdd(&lds_sq[j],  accsq);
    __syncthreads();
    if (tid < DIM) {
        atomicAdd(&ssum[tid], lds_sum[tid]);
        atomicAdd(&ssq[tid],  lds_sq[tid]);
    }
}

__global__ void finalize_stats(const float* __restrict__ ssum,
                               const float* __restrict__ ssq,
                               float* __restrict__ mu,
                               float* __restrict__ rsig) {
    int jf = threadIdx.x;   // 128 threads
    float m   = ssum[jf] * (1.0f / (float)N_NODES);
    float var = ssq[jf] * (1.0f / (float)N_NODES) - m * m;
    mu[jf]   = m;
    rsig[jf] = rsqrtf(var + BN_EPS);
}

// in-place batchnorm on the z_cat slice + global_add_pool into g_cat slice
__global__ void bn_pool(float* __restrict__ h,        // z_cat + layer col, stride 384
                        const float* __restrict__ mu,
                        const float* __restrict__ rsig,
                        const float* __restrict__ gamma,
                        const float* __restrict__ beta,
                        const int* __restrict__ batch,
                        float* __restrict__ g) {      // g_cat + layer col, stride 384
    int t = blockIdx.x * blockDim.x + threadIdx.x;    // < N_NODES * 128
    int row = t >> 7;
    int jf  = t & 127;
    size_t off = (size_t)row * OUT_DIM + jf;
    float val = (h[off] - mu[jf]) * rsig[jf] * gamma[jf] + beta[jf];
    h[off] = val;
    atomicAdd(&g[(size_t)batch[row] * OUT_DIM + jf], val);
}

// ---------------------------------------------------------------- launcher
extern "C" void kernel_launch(void* const* d_in, const int* in_sizes, int n_in,
                              void* d_out, int out_size, void* d_ws, size_t ws_size,
                              hipStream_t stream) {
    const float* x      = (const float*)d_in[0];
    const float* W      = (const float*)d_in[1];   // [3][128][128]
    const float* b      = (const float*)d_in[2];   // [3][128]
    const float* eps_p  = (const float*)d_in[3];   // [3]
    const float* gamma  = (const float*)d_in[4];   // [3][128]
    const float* beta   = (const float*)d_in[5];   // [3][128]
    const int*   ei     = (const int*)d_in[6];     // [2][N_EDGES]
    const int*   batch  = (const int*)d_in[7];     // [N_NODES]

    float* zcat = (float*)d_out;                           // [N_NODES][384]
    float* gcat = zcat + (size_t)N_NODES * OUT_DIM;        // [512][384]

    float* u    = (float*)d_ws;                            // [N_NODES][128]
    float* ssum = u + (size_t)N_NODES * DIM;               // [128]
    float* ssq  = ssum + DIM;                              // [128]
    float* mu   = ssq + DIM;                               // [128]
    float* rsig = mu + DIM;                                // [128]

    // zero graph pool output
    {
        int n = N_GRAPHS * OUT_DIM;
        zero_f32<<<(n + 255) / 256, 256, 0, stream>>>(gcat, n);
    }

    for (int i = 0; i < N_LAYERS; ++i) {
        const float* z = (i == 0) ? x : (zcat + (size_t)(i - 1) * DIM);
        int zstride    = (i == 0) ? DIM : OUT_DIM;

        // u = (1+eps_i) * z
        init_u<<<(N_NODES * 32) / 256, 256, 0, stream>>>(u, z, zstride, eps_p + i);
        // zero BN accumulators (ssum and ssq are contiguous 256 floats)
        zero_f32<<<1, 256, 0, stream>>>(ssum, 2 * DIM);
        // u[dst] += z[src] over all edges
        scatter_edges<<<(N_EDGES * 32) / 256, 256, 0, stream>>>(
            u, z, zstride, ei, ei + N_EDGES);
        // h = relu(u @ W_i^T + b_i) into z_cat slice; BN stats accumulated
        // 6250 strips of 16 rows; 625 blocks x 10 strips
        gemm_bias_relu_stats<<<625, 256, 0, stream>>>(
            u, W + (size_t)i * DIM * DIM, b + (size_t)i * DIM,
            zcat + (size_t)i * DIM, ssum, ssq, 10);
        finalize_stats<<<1, DIM, 0, stream>>>(ssum, ssq, mu, rsig);
        // in-place BN + pooled sums
        bn_pool<<<(N_NODES * DIM) / 256, 256, 0, stream>>>(
            zcat + (size_t)i * DIM, mu, rsig,
            gamma + (size_t)i * DIM, beta + (size_t)i * DIM,
            batch, gcat + (size_t)i * DIM);
    }
}